// WindowWarp_81389630259938
// MI455X (gfx1250) — compile-verified
//
#include <hip/hip_runtime.h>
#include <hip/hip_bf16.h>

// WindowWarp, fused single-pass 4-tap gather.
//   B=256, T=2048, C=64, WARP_SIZE=ceil(0.1*2048)=205, SCALES={0.5,2.0}
//   L_MAX = 2048 - 205 + int(205*2.0) = 2253
// DRAM-bound op: ~256 MiB total traffic -> ~11us floor @ 23.3 TB/s.
// Key MI455X lever: NT stores keep the 128 MiB gather source resident in the
// 192 MiB L2; loads are 128-bit (global_load_b128), stores are nt-hinted
// global_store_b128.

#define B_LEN 256
#define T_LEN 2048
#define C_LEN 64
#define WSZ   205
#define L_MAXC 2253

typedef float v4f __attribute__((ext_vector_type(4)));

// One warped-row sample: warped[b, j, 4*cq .. 4*cq+3]
// = lerp over x rows at s(j), replicating reference arithmetic exactly.
__device__ __forceinline__ v4f warp_row_gather(const float* __restrict__ xb,
                                               int j, float start, float nl,
                                               int cq) {
    const float jf   = (float)j;
    const float send = start + nl;                       // window end (float)
    const float denom = fmaxf(nl - 1.0f, 1.0f);
    // reference: start + (j - start) * (WARP_SIZE - 1) / max(nl - 1, 1)
    const float s_win   = start + (jf - start) * (float)(WSZ - 1) / denom;
    const float s_after = jf - nl + (float)WSZ;
    float s = (jf >= start && jf < send) ? s_win
            : ((jf >= send) ? s_after : jf);
    s = fminf(fmaxf(s, 0.0f), (float)(T_LEN - 1));       // clip to [0, T-1]

    const float q0 = floorf(s);
    const float fs = s - q0;                             // frac
    const int k0 = (int)q0;
    const int k1 = min(k0 + 1, T_LEN - 1);

    const v4f a = ((const v4f*)(xb + (size_t)k0 * C_LEN))[cq];
    const v4f c = ((const v4f*)(xb + (size_t)k1 * C_LEN))[cq];
    return a * (1.0f - fs) + c * fs;                     // g0*(1-f) + g1*f
}

__global__ __launch_bounds__(256) void window_warp_kernel(
    const float* __restrict__ x,            // [B, T, C]
    const float* __restrict__ warp_scales,  // [B]
    const int*   __restrict__ window_starts,// [B]
    float*       __restrict__ out)          // [B, T, C]
{
    const int cq = threadIdx.x & 15;        // which float4 of the 16 in a row
    const int tr = threadIdx.x >> 4;        // row-in-tile 0..15
    const int t  = blockIdx.x * 16 + tr;    // output time index
    const int b  = blockIdx.y;              // uniform per block -> scalar loads

    const float scale = warp_scales[b];
    const float start = (float)window_starts[b];
    const float nl = floorf((float)WSZ * scale);         // new_len (float)
    const float L  = (float)(T_LEN - WSZ) + nl;          // 1945 or 2253

    // Second-stage resample index: u = t*(L-1)/(T-1), clipped.
    float u = (float)t * (L - 1.0f) / (float)(T_LEN - 1);
    u = fminf(fmaxf(u, 0.0f), L - 1.0f);
    const float p0 = floorf(u);
    const float fu = u - p0;
    const int i0 = (int)p0;
    const int i1 = min(i0 + 1, L_MAXC - 1);

    const float* xb = x + (size_t)b * T_LEN * C_LEN;
    const v4f w0 = warp_row_gather(xb, i0, start, nl, cq);
    const v4f w1 = warp_row_gather(xb, i1, start, nl, cq);
    const v4f o  = w0 * (1.0f - fu) + w1 * fu;

    // Write-once output: non-temporal so x stays resident in the 192 MiB L2.
    v4f* dst = (v4f*)(out + ((size_t)b * T_LEN + t) * C_LEN) + cq;
    __builtin_nontemporal_store(o, dst);
}

extern "C" void kernel_launch(void* const* d_in, const int* in_sizes, int n_in,
                              void* d_out, int out_size, void* d_ws, size_t ws_size,
                              hipStream_t stream) {
    (void)in_sizes; (void)n_in; (void)d_ws; (void)ws_size; (void)out_size;
    const float* x      = (const float*)d_in[0];
    const float* scales = (const float*)d_in[1];
    const int*   starts = (const int*)d_in[2];
    float*       out    = (float*)d_out;

    dim3 grid(T_LEN / 16, B_LEN, 1);   // 128 x 256 blocks
    dim3 block(256, 1, 1);             // 8 wave32 per block
    window_warp_kernel<<<grid, block, 0, stream>>>(x, scales, starts, out);
}